// MatMulEmbed_34651796144318
// MI455X (gfx1250) — compile-verified
//
#include <hip/hip_runtime.h>
#include <stdint.h>

// Embedding gather: out[row, :] = embedding[tokens[row], :]
// (mathematically identical to one_hot(tokens) @ embedding)
//
// MI455X strategy: this op is pure data movement (~32 MB @ 23.3 TB/s ~ 1.4 us);
// matrix cores are useless here. Use the CDNA5 async DMA path:
//   global_load_async_to_lds_b128  (ASYNCcnt, fire-and-forget, no VGPR round trip)
//   s_wait_asynccnt 0
//   global_store_async_from_lds_b128
// Each lane moves 16B per row; 256 lanes == one 4KB row (1024 f32).
// 4 rows per block -> 4 outstanding DMA requests per lane before the wait.

#define FEATURES        1024
#define ROWS_PER_BLOCK  4
#define THREADS         256   // 8 wave32 waves

__global__ __launch_bounds__(THREADS)
void MatMulEmbed_gather_kernel(const int* __restrict__ tokens,
                               const float* __restrict__ emb,
                               float* __restrict__ out,
                               int n_rows) {
  __shared__ float lds[ROWS_PER_BLOCK * FEATURES];   // 16 KB

  const int tid  = threadIdx.x;
  const int col  = tid * 4;                          // 4 floats = 16B per lane
  const int row0 = blockIdx.x * ROWS_PER_BLOCK;

  // Issue all async DMA loads (global -> LDS) before any wait: 4 in flight/lane.
#pragma unroll
  for (int r = 0; r < ROWS_PER_BLOCK; ++r) {
    const int row = row0 + r;
    if (row < n_rows) {                              // uniform per block
      const int tok = tokens[row];                   // broadcast load
      const float* gsrc = emb + (size_t)tok * FEATURES + col;
      // low 32 bits of a flat LDS pointer == wave-relative LDS byte address
      const unsigned lds_addr =
          (unsigned)(uintptr_t)(const void*)&lds[r * FEATURES + col];
      asm volatile("global_load_async_to_lds_b128 %0, %1, off"
                   :
                   : "v"(lds_addr), "v"(gsrc)
                   : "memory");
    }
  }

  // Each lane only reads back the LDS bytes it wrote itself, so no barrier is
  // needed — just drain the async loads before issuing the async stores.
  asm volatile("s_wait_asynccnt 0" ::: "memory");

#pragma unroll
  for (int r = 0; r < ROWS_PER_BLOCK; ++r) {
    const int row = row0 + r;
    if (row < n_rows) {
      float* gdst = out + (size_t)row * FEATURES + col;
      const unsigned lds_addr =
          (unsigned)(uintptr_t)(const void*)&lds[r * FEATURES + col];
      asm volatile("global_store_async_from_lds_b128 %0, %1, off"
                   :
                   : "v"(gdst), "v"(lds_addr)
                   : "memory");
    }
  }
  // Async stores drain via the implicit wait-idle at s_endpgm.
}

extern "C" void kernel_launch(void* const* d_in, const int* in_sizes, int n_in,
                              void* d_out, int out_size, void* d_ws, size_t ws_size,
                              hipStream_t stream) {
  const int*   tokens = (const int*)d_in[0];     // (2, 2048) int32 -> 4096 rows
  const float* emb    = (const float*)d_in[1];   // (32000, 1024) f32
  float*       out    = (float*)d_out;           // (2, 2048, 1024) f32

  const int n_rows = in_sizes[0];                // 4096
  const int blocks = (n_rows + ROWS_PER_BLOCK - 1) / ROWS_PER_BLOCK;

  MatMulEmbed_gather_kernel<<<blocks, THREADS, 0, stream>>>(tokens, emb, out, n_rows);
}